// Decoder_61366492725853
// MI455X (gfx1250) — compile-verified
//
#include <hip/hip_runtime.h>
#include <hip/hip_bf16.h>
#include <math.h>

// ---------------------------------------------------------------------------
// Pointer-network greedy decoder for MI455X (gfx1250, wave32).
//  B=512, L=128, D=128. Batch rows independent -> persistent per-tile decode.
//  64 WGs x 512 threads, 8 batch rows per WG (WMMA M=16 half-used; GEMM is a
//  small fraction of step time, logit phases dominate and need WGP spread).
//  GEMMs: exact-fp32 V_WMMA_F32_16X16X4_F32. tanh: hardware V_TANH_F32.
//  Working set (~130MB: enc+emb+ref_p+ref_g+weights) fits the 192MB L2.
// ---------------------------------------------------------------------------

#define B_ 512
#define L_ 128
#define D_ 128
#define STEPS (L_ - 2)
#define CCLIP 10.0f
#define LDA 132          // padded LDS row stride for WMMA A operands
#define ROWS 8           // batch rows per workgroup
#define NTHREADS 512     // 16 waves

typedef float v2f __attribute__((ext_vector_type(2)));
typedef float v8f __attribute__((ext_vector_type(8)));

// Hardware transcendental tanh (gfx1250 V_TANH_F32). ISA 7.4: one V_NOP (or
// independent op) after a TRANS op before its output is consumed.
__device__ __forceinline__ float htanh(float x) {
  float y;
  asm volatile("v_tanh_f32 %0, %1\n\tv_nop" : "=v"(y) : "v"(x));
  return y;
}
__device__ __forceinline__ float hsigm(float x) {
  return 0.5f * htanh(0.5f * x) + 0.5f;
}

// C(16x16) += A(16xK, row-major lda) * W^T   where W is row-major [N][K].
// CDNA5 16x4 f32 A layout: lanes 0-15 hold (K=k,k+1); lanes 16-31 (K=k+2,k+3);
// B mirrors with N across lanes. C/D: VGPR r -> M=r / r+8 per lane half.
__device__ __forceinline__ v8f wmma_a_x_wT(const float* __restrict__ A, int lda,
                                           const float* __restrict__ W, int ldw,
                                           int K, v8f acc) {
  const int lane = threadIdx.x & 31;
  const int half = lane >> 4;
  const int r    = lane & 15;
  const float* arow = A + r * lda + 2 * half;
  const float* wrow = W + r * ldw + 2 * half;
  for (int k = 0; k < K; k += 4) {
    v2f a, b;
    a[0] = arow[k]; a[1] = arow[k + 1];
    b[0] = wrow[k]; b[1] = wrow[k + 1];
    acc = __builtin_amdgcn_wmma_f32_16x16x4_f32(false, a, false, b,
                                                (short)0, acc, false, false);
  }
  return acc;
}

// ---------------------------------------------------------------------------
// Kernel 1: ref_p = enc @ wr_p^T + br_p ; ref_g = enc @ wr_g^T + br_g
// enc is (L,B,D) flattened -> 65536 rows of 128. 16 rows per block.
// ---------------------------------------------------------------------------
__global__ __launch_bounds__(256) void ref_precompute(
    const float* __restrict__ enc,
    const float* __restrict__ wr_p, const float* __restrict__ br_p,
    const float* __restrict__ wr_g, const float* __restrict__ br_g,
    float* __restrict__ ref_p, float* __restrict__ ref_g) {
  __shared__ float sA[16 * LDA];
  const int tid  = threadIdx.x;
  const int wave = tid >> 5;
  const int lane = tid & 31;
  const int half = lane >> 4;
  const int ln   = lane & 15;
  const long rowbase = (long)blockIdx.x * 16;

  for (int i = tid; i < 16 * D_; i += 256) {
    int m = i >> 7, d = i & 127;
    sA[m * LDA + d] = enc[rowbase * D_ + i];  // 16 rows are contiguous
  }
  __syncthreads();

  const int nt = wave;  // 8 waves <-> 8 N-tiles of 16
  v8f accp = {0.f,0.f,0.f,0.f,0.f,0.f,0.f,0.f};
  v8f accg = {0.f,0.f,0.f,0.f,0.f,0.f,0.f,0.f};
  accp = wmma_a_x_wT(sA, LDA, wr_p + nt * 16 * D_, D_, D_, accp);
  accg = wmma_a_x_wT(sA, LDA, wr_g + nt * 16 * D_, D_, D_, accg);

  const int n = nt * 16 + ln;
  const float bp = br_p[n], bg = br_g[n];
  for (int vr = 0; vr < 8; ++vr) {
    int m = vr + 8 * half;
    ref_p[(rowbase + m) * D_ + n] = accp[vr] + bp;
    ref_g[(rowbase + m) * D_ + n] = accg[vr] + bg;
  }
}

// ---------------------------------------------------------------------------
// Kernel 2: persistent decode. 64 blocks x 512 threads; block owns 8 rows.
// WMMA uses 16-row tiles; rows 8-15 of A are zero or garbage -> they only
// affect D rows 8-15, which are never stored.
// ---------------------------------------------------------------------------
__global__ __launch_bounds__(NTHREADS) void decode_kernel(
    const float* __restrict__ dec0, const float* __restrict__ emb,
    const float* __restrict__ h0,   const float* __restrict__ c0,
    const float* __restrict__ enc,
    const float* __restrict__ w_ih, const float* __restrict__ w_hh,
    const float* __restrict__ b_ih, const float* __restrict__ b_hh,
    const float* __restrict__ wq_p, const float* __restrict__ bq_p,
    const float* __restrict__ v_p,
    const float* __restrict__ wq_g, const float* __restrict__ bq_g,
    const float* __restrict__ v_g,
    const float* __restrict__ ref_p, const float* __restrict__ ref_g,
    float* __restrict__ out_probs, float* __restrict__ out_idx) {
  const int tid  = threadIdx.x;
  const int wave = tid >> 5;
  const int lane = tid & 31;
  const int half = lane >> 4;
  const int ln   = lane & 15;
  const int bb0  = blockIdx.x * ROWS;  // batch tile base

  __shared__ float sX[16 * LDA];       // dec_in (rows 8-15 stay zero)
  __shared__ float sH[16 * LDA];       // hidden (rows 8-15 stay zero)
  __shared__ float sC[16 * LDA];       // cell
  __shared__ float sGates[16 * 512];   // aliased by q/u/g after LSTM consumes
  __shared__ float sVg[D_], sVp[D_];
  __shared__ unsigned int sMask[ROWS][4];
  __shared__ int   sIdx[ROWS];
  __shared__ float sRedV[ROWS][32];
  __shared__ int   sRedI[ROWS][32];

  float* sQ = sGates;                  // 16 x 128 region (rows >=8 scratch)
  float* sU = sGates + 16 * 128;       // 8 x 128 used
  float* sG = sGates + 32 * 128;       // 16 x LDA region (WMMA A source)

  // ---- init ----
  for (int i = tid; i < 16 * LDA; i += NTHREADS) {
    sX[i] = 0.f; sH[i] = 0.f; sC[i] = 0.f;
  }
  __syncthreads();
  for (int i = tid; i < ROWS * D_; i += NTHREADS) {
    int m = i >> 7, d = i & 127;
    sX[m * LDA + d] = dec0[(bb0 + m) * D_ + d];
    sH[m * LDA + d] = h0[(bb0 + m) * D_ + d];
    sC[m * LDA + d] = c0[(bb0 + m) * D_ + d];
  }
  if (tid < D_) { sVg[tid] = v_g[tid]; sVp[tid] = v_p[tid]; }
  if (tid < ROWS) {
    sMask[tid][0] = 3u;  // positions 0,1 pre-visited
    sMask[tid][1] = 0u; sMask[tid][2] = 0u; sMask[tid][3] = 0u;
  }
  __syncthreads();

  for (int t = 0; t < STEPS; ++t) {
    // ---- LSTM gates: gates = X@w_ih^T + H@w_hh^T + biases (rows 0-7) ----
    for (int nt = wave; nt < 32; nt += 16) {   // 16 waves, 2 tiles each
      v8f acc = {0.f,0.f,0.f,0.f,0.f,0.f,0.f,0.f};
      acc = wmma_a_x_wT(sX, LDA, w_ih + nt * 16 * D_, D_, D_, acc);
      acc = wmma_a_x_wT(sH, LDA, w_hh + nt * 16 * D_, D_, D_, acc);
      if (half == 0) {                         // only M rows 0-7 are real
        const int n = nt * 16 + ln;
        const float bias = b_ih[n] + b_hh[n];
        for (int vr = 0; vr < 8; ++vr)
          sGates[vr * 512 + n] = acc[vr] + bias;
      }
    }
    __syncthreads();

    // ---- LSTM cell update (gate order i,f,g,o) ----
    for (int i = tid; i < ROWS * D_; i += NTHREADS) {
      int m = i >> 7, d = i & 127;
      float gi = sGates[m * 512 + d];
      float gf = sGates[m * 512 + 128 + d];
      float gg = sGates[m * 512 + 256 + d];
      float go = sGates[m * 512 + 384 + d];
      float c2 = hsigm(gf) * sC[m * LDA + d] + hsigm(gi) * htanh(gg);
      sC[m * LDA + d] = c2;
      sH[m * LDA + d] = hsigm(go) * htanh(c2);
    }
    __syncthreads();

    // ---- glimpse query: q = H @ wq_g^T + bq_g ----
    if (wave < 8) {
      const int nt = wave;
      v8f acc = {0.f,0.f,0.f,0.f,0.f,0.f,0.f,0.f};
      acc = wmma_a_x_wT(sH, LDA, wq_g + nt * 16 * D_, D_, D_, acc);
      if (half == 0) {
        const int n = nt * 16 + ln;
        const float bias = bq_g[n];
        for (int vr = 0; vr < 8; ++vr)
          sQ[vr * 128 + n] = acc[vr] + bias;
      }
    }
    __syncthreads();

    // ---- glimpse logits: u[m][l] = v_g . tanh(ref_g[l,b] + q[m]) ----
    for (int i = tid; i < ROWS * L_; i += NTHREADS) {
      int m = i >> 7, l = i & 127;
      bool masked = (t > 0) && ((sMask[m][l >> 5] >> (l & 31)) & 1u);
      float u;
      if (masked) {
        u = -__builtin_inff();
      } else {
        const float4* rg4 = (const float4*)(ref_g + ((long)l * B_ + bb0 + m) * D_);
        const float4* qr4 = (const float4*)(sQ + m * 128);
        const float4* vv4 = (const float4*)sVg;
        float s = 0.f;
        for (int d4 = 0; d4 < D_ / 4; ++d4) {
          float4 r = rg4[d4], q = qr4[d4], vv = vv4[d4];
          s += vv.x * htanh(r.x + q.x);
          s += vv.y * htanh(r.y + q.y);
          s += vv.z * htanh(r.z + q.z);
          s += vv.w * htanh(r.w + q.w);
        }
        u = s;
      }
      sU[m * 128 + l] = u;
    }
    __syncthreads();

    // ---- glimpse softmax (in place in sU): 32 threads per row ----
    {
      const bool act = tid < ROWS * 32;
      const int m = tid >> 5, j = tid & 31;
      float ev[4];
      if (act) {
        float mx = -__builtin_inff();
        for (int k = 0; k < 4; ++k) mx = fmaxf(mx, sU[m * 128 + j + 32 * k]);
        sRedV[m][j] = mx;
      }
      __syncthreads();
      float rowmax = -__builtin_inff();
      if (act)
        for (int jj = 0; jj < 32; ++jj) rowmax = fmaxf(rowmax, sRedV[m][jj]);
      __syncthreads();
      float ps = 0.f;
      if (act) {
        for (int k = 0; k < 4; ++k) {
          float e = __expf(sU[m * 128 + j + 32 * k] - rowmax);
          ev[k] = e; ps += e;
        }
        sRedV[m][j] = ps;
      }
      __syncthreads();
      if (act) {
        float rs = 0.f;
        for (int jj = 0; jj < 32; ++jj) rs += sRedV[m][jj];
        float inv = 1.f / rs;
        for (int k = 0; k < 4; ++k) sU[m * 128 + j + 32 * k] = ev[k] * inv;
      }
    }
    __syncthreads();

    // ---- glimpse readout: g[m][d] = sum_l p[m][l] * enc[l,b,d] ----
    for (int i = tid; i < ROWS * D_; i += NTHREADS) {
      int m = i >> 7, d = i & 127;
      float s = 0.f;
      for (int l = 0; l < L_; ++l)
        s += sU[m * 128 + l] * enc[((long)l * B_ + bb0 + m) * D_ + d];
      sG[m * LDA + d] = s;
    }
    __syncthreads();

    // ---- pointer query: q = G @ wq_p^T + bq_p (overwrites sQ) ----
    if (wave < 8) {
      const int nt = wave;
      v8f acc = {0.f,0.f,0.f,0.f,0.f,0.f,0.f,0.f};
      acc = wmma_a_x_wT(sG, LDA, wq_p + nt * 16 * D_, D_, D_, acc);
      if (half == 0) {
        const int n = nt * 16 + ln;
        const float bias = bq_p[n];
        for (int vr = 0; vr < 8; ++vr)
          sQ[vr * 128 + n] = acc[vr] + bias;
      }
    }
    __syncthreads();

    // ---- pointer logits: u = C*tanh(v_p . tanh(ref_p + q)), then mask ----
    for (int i = tid; i < ROWS * L_; i += NTHREADS) {
      int m = i >> 7, l = i & 127;
      bool masked = (t > 0) && ((sMask[m][l >> 5] >> (l & 31)) & 1u);
      float u;
      if (masked) {
        u = -__builtin_inff();
      } else {
        const float4* rp4 = (const float4*)(ref_p + ((long)l * B_ + bb0 + m) * D_);
        const float4* qr4 = (const float4*)(sQ + m * 128);
        const float4* vv4 = (const float4*)sVp;
        float s = 0.f;
        for (int d4 = 0; d4 < D_ / 4; ++d4) {
          float4 r = rp4[d4], q = qr4[d4], vv = vv4[d4];
          s += vv.x * htanh(r.x + q.x);
          s += vv.y * htanh(r.y + q.y);
          s += vv.z * htanh(r.z + q.z);
          s += vv.w * htanh(r.w + q.w);
        }
        u = CCLIP * htanh(s);
      }
      sU[m * 128 + l] = u;
    }
    __syncthreads();

    // ---- pointer softmax + write probs + greedy argmax + mask update ----
    {
      const bool act = tid < ROWS * 32;
      const int m = tid >> 5, j = tid & 31;
      float ev[4];
      float rowmax = -__builtin_inff(); int rowidx = 0;
      if (act) {
        float mx = -__builtin_inff(); int mi = j;
        for (int k = 0; k < 4; ++k) {
          int l = j + 32 * k;
          float v = sU[m * 128 + l];
          if (v > mx) { mx = v; mi = l; }
        }
        sRedV[m][j] = mx; sRedI[m][j] = mi;
      }
      __syncthreads();
      if (act) {
        for (int jj = 0; jj < 32; ++jj) {
          float v = sRedV[m][jj];
          if (v > rowmax) { rowmax = v; rowidx = sRedI[m][jj]; }
        }
      }
      __syncthreads();
      if (act) {
        float ps = 0.f;
        for (int k = 0; k < 4; ++k) {
          float e = __expf(sU[m * 128 + j + 32 * k] - rowmax);
          ev[k] = e; ps += e;
        }
        sRedV[m][j] = ps;
      }
      __syncthreads();
      if (act) {
        float rs = 0.f;
        for (int jj = 0; jj < 32; ++jj) rs += sRedV[m][jj];
        float inv = 1.f / rs;
        float* orow = out_probs + ((long)t * B_ + bb0 + m) * L_;
        for (int k = 0; k < 4; ++k) orow[j + 32 * k] = ev[k] * inv;
        if (j == 0) {
          sIdx[m] = rowidx;
          sMask[m][rowidx >> 5] |= 1u << (rowidx & 31);
          out_idx[(long)t * B_ + bb0 + m] = (float)rowidx;
        }
      }
    }
    __syncthreads();

    // ---- next decoder input: dec_in = emb[idx, b, :] ----
    for (int i = tid; i < ROWS * D_; i += NTHREADS) {
      int m = i >> 7, d = i & 127;
      sX[m * LDA + d] = emb[((long)sIdx[m] * B_ + bb0 + m) * D_ + d];
    }
    __syncthreads();
  }
}

// ---------------------------------------------------------------------------
extern "C" void kernel_launch(void* const* d_in, const int* in_sizes, int n_in,
                              void* d_out, int out_size, void* d_ws, size_t ws_size,
                              hipStream_t stream) {
  const float* decoder_input = (const float*)d_in[0];
  const float* embedding_x   = (const float*)d_in[1];
  const float* h0    = (const float*)d_in[2];
  const float* c0    = (const float*)d_in[3];
  const float* enc   = (const float*)d_in[4];
  const float* w_ih  = (const float*)d_in[5];
  const float* w_hh  = (const float*)d_in[6];
  const float* b_ih  = (const float*)d_in[7];
  const float* b_hh  = (const float*)d_in[8];
  const float* wq_p  = (const float*)d_in[9];
  const float* bq_p  = (const float*)d_in[10];
  const float* wr_p  = (const float*)d_in[11];
  const float* br_p  = (const float*)d_in[12];
  const float* v_p   = (const float*)d_in[13];
  const float* wq_g  = (const float*)d_in[14];
  const float* bq_g  = (const float*)d_in[15];
  const float* wr_g  = (const float*)d_in[16];
  const float* br_g  = (const float*)d_in[17];
  const float* v_g   = (const float*)d_in[18];

  float* out   = (float*)d_out;
  float* ref_p = (float*)d_ws;                       // 32 MB
  float* ref_g = ref_p + (size_t)L_ * B_ * D_;       // 32 MB

  hipLaunchKernelGGL(ref_precompute, dim3(L_ * B_ / 16), dim3(256), 0, stream,
                     enc, wr_p, br_p, wr_g, br_g, ref_p, ref_g);

  hipLaunchKernelGGL(decode_kernel, dim3(B_ / ROWS), dim3(NTHREADS), 0, stream,
                     decoder_input, embedding_x, h0, c0, enc,
                     w_ih, w_hh, b_ih, b_hh,
                     wq_p, bq_p, v_p, wq_g, bq_g, v_g,
                     ref_p, ref_g,
                     out, out + (size_t)STEPS * B_ * L_);
}